// Despawn_15324443312360
// MI455X (gfx1250) — compile-verified
//
#include <hip/hip_runtime.h>
#include <stdint.h>

#define LEVELS 10
#define ROW    65536
#define TPB    1024

#if __has_builtin(__builtin_amdgcn_global_load_async_to_lds_b128)
#define HAVE_ASYNC_LDS 1
#else
#define HAVE_ASYNC_LDS 0
#endif

#if __has_builtin(__builtin_amdgcn_wmma_f32_16x16x4_f32)
#define HAVE_WMMA_F32 1
#else
#define HAVE_WMMA_F32 0
#endif

typedef int   v4i __attribute__((ext_vector_type(4)));
typedef float v2f __attribute__((ext_vector_type(2)));
typedef float v8f __attribute__((ext_vector_type(8)));
typedef __attribute__((address_space(1))) v4i* gv4i_ptr;   // global int4*
typedef __attribute__((address_space(3))) v4i* lv4i_ptr;   // LDS int4*

__device__ __forceinline__ float sigmf(float t) {
  return 1.0f / (1.0f + __expf(-t));
}

__device__ __forceinline__ float thresh(float v, float alpha, float bp, float bm) {
  return v * (sigmf(alpha * (v - bp)) + sigmf(-alpha * (v + bm)));
}

// wav[j]  = flip(s_rec)[j] * (j odd  ? -1 : +1)
__device__ __forceinline__ void ld_fwd_filters(const float* sc, const float* sr, int lvl,
                                               float* sf, float* wf) {
#pragma unroll
  for (int k = 0; k < 8; k++) {
    sf[k] = sc[lvl * 8 + k];
    float g = sr[lvl * 8 + 7 - k];
    wf[k] = (k & 1) ? -g : g;
  }
}

// ws[j]   = flip(s_rec)[j] * (j even ? -1 : +1)
__device__ __forceinline__ void ld_inv_filters(const float* sc, const float* sr, int lvl,
                                               float* sf, float* wsf) {
#pragma unroll
  for (int k = 0; k < 8; k++) {
    sf[k] = sc[lvl * 8 + k];
    float g = sr[lvl * 8 + 7 - k];
    wsf[k] = (k & 1) ? g : -g;
  }
}

// ---------------- VALU lifting (small levels) ----------------

template <int C>
__device__ void fwd_level(float* a, int H, const float* wf, const float* sf,
                          float alpha, float bp, float bm, float& regsum, int tid) {
  constexpr int CH = (C < 8) ? C : 8;
  const int mask = H - 1;
  const int n0 = tid * C;
  const bool active = (C > 1) || (tid < H);
  float oddU[C], evenU[C];
  if (active) {
#pragma unroll
    for (int jc = 0; jc < C; jc += CH) {
      float we[CH + 7], wo[CH + 7];
#pragma unroll
      for (int m = 0; m < CH + 7; m++) {
        int i = (n0 + jc - 7 + m) & mask;
        we[m] = a[2 * i];
        wo[m] = a[2 * i + 1];
      }
#pragma unroll
      for (int j = 0; j < CH; j++) {
        float ao = wo[j + 7];
        float ae = we[j + 7];
#pragma unroll
        for (int k = 0; k < 8; k++) {
          ao -= wf[k] * we[j + 7 - k];  // odd  - circconv(even, wav)
          ae -= sf[k] * wo[j + 7 - k];  // even - circconv(odd,  scaling)
        }
        oddU[jc + j] = ao;
        evenU[jc + j] = ae;
      }
    }
  }
  __syncthreads();  // all reads done before in-place writes
  if (active) {
#pragma unroll
    for (int j = 0; j < C; j++) {
      a[n0 + j] = oddU[j];
      float t = thresh(evenU[j], alpha, bp, bm);
      a[H + n0 + j] = t;
      regsum += fabsf(t);
    }
  }
  __syncthreads();
}

template <int C>
__device__ void inv_level(float* a, int H, const float* sf, const float* wsf, int tid) {
  constexpr int CH = (C < 8) ? C : 8;
  const int mask = H - 1;
  const int n0 = tid * C;
  const bool active = (C > 1) || (tid < H);
  float er[C], orr[C];
  if (active) {
#pragma unroll
    for (int jc = 0; jc < C; jc += CH) {
      float wd[CH + 7], wa[CH + 7];
#pragma unroll
      for (int m = 0; m < CH + 7; m++) {
        int i = (n0 + jc + m) & mask;
        wd[m] = a[H + i];
        wa[m] = a[i];
      }
#pragma unroll
      for (int j = 0; j < CH; j++) {
        float ae = wd[j];
        float ao = wa[j];
#pragma unroll
        for (int k = 0; k < 8; k++) {
          ae += sf[k]  * wa[j + k];  // details + corr(approx, scaling)
          ao += wsf[k] * wd[j + k];  // approx  + corr(details, ws)
        }
        er[jc + j] = ae;
        orr[jc + j] = ao;
      }
    }
  }
  __syncthreads();
  if (active) {
#pragma unroll
    for (int j = 0; j < C; j++) {
      int n = n0 + j;
      a[2 * n]     = er[j];
      a[2 * n + 1] = orr[j];
    }
  }
  __syncthreads();
}

// ---------------- WMMA lifting (H >= 8192) ----------------
// Per wave: 256 outputs/parity per batch as C[16x16] = A[16x24] * Toeplitz[24x16],
// K=24 swept by 6 chained V_WMMA_F32_16X16X4_F32.
// A 16x4 f32 layout: lanes 0-15 -> K=0,1 ; lanes 16-31 -> K=2,3 ; M = lane&15.
// B 4x16 layout: vgpr v -> row K=v (lanes 0-15) / K=v+2 (lanes 16-31), N = lane&15.
// C/D 16x16: vgpr v -> M = v + 8*(lane>>4), N = lane&15.

#if HAVE_WMMA_F32

template <int NB>
__device__ void fwd_level_wmma(float* a, int H, const float* bt0, const float* bt1,
                               float alpha, float bp, float bm, float& regsum, int tid) {
  const int mask = H - 1;
  const int wave = tid >> 5;
  const int lane = tid & 31;
  const int lo = lane & 15;
  const int hi = lane >> 4;
  const int kloc = hi ? 2 : 0;

  // Hoist Toeplitz bands into registers (bt0: wav for odd_u, bt1: scaling for even_u)
  v2f Bw[6], Bs[6];
#pragma unroll
  for (int s = 0; s < 6; s++) {
    int r = 4 * s + kloc;
    Bw[s] = v2f{bt0[r * 16 + lo], bt0[(r + 1) * 16 + lo]};
    Bs[s] = v2f{bt1[r * 16 + lo], bt1[(r + 1) * 16 + lo]};
  }

  v8f accO[NB], accE[NB];
#pragma unroll
  for (int b = 0; b < NB; b++) {
    const int n0 = (wave + 32 * b) * 256;
    const int base = n0 + 16 * lo - 7;  // window start for A-row M = lo
    v8f aco = {0.f, 0.f, 0.f, 0.f, 0.f, 0.f, 0.f, 0.f};
    v8f ace = {0.f, 0.f, 0.f, 0.f, 0.f, 0.f, 0.f, 0.f};
#pragma unroll
    for (int s = 0; s < 6; s++) {
      int m0 = 4 * s + kloc;
      int i0 = (base + m0) & mask;
      int i1 = (base + m0 + 1) & mask;
      v2f Ae = v2f{a[2 * i0],     a[2 * i1]};      // even window
      v2f Ao = v2f{a[2 * i0 + 1], a[2 * i1 + 1]};  // odd window
      aco = __builtin_amdgcn_wmma_f32_16x16x4_f32(false, Ae, false, Bw[s],
                                                  (short)0, aco, false, false);
      ace = __builtin_amdgcn_wmma_f32_16x16x4_f32(false, Ao, false, Bs[s],
                                                  (short)0, ace, false, false);
    }
    // fold pass-through terms while still in the read phase
    v8f ov, ev;
#pragma unroll
    for (int v = 0; v < 8; v++) {
      int n = n0 + 16 * (v + 8 * hi) + lo;
      ov[v] = a[2 * n + 1];
      ev[v] = a[2 * n];
    }
    accO[b] = ov - aco;  // odd_u  = odd  - conv(even, wav)
    accE[b] = ev - ace;  // even_u = even - conv(odd,  scaling)
  }
  __syncthreads();
#pragma unroll
  for (int b = 0; b < NB; b++) {
    const int n0 = (wave + 32 * b) * 256;
#pragma unroll
    for (int v = 0; v < 8; v++) {
      int n = n0 + 16 * (v + 8 * hi) + lo;
      a[n] = accO[b][v];
      float t = thresh(accE[b][v], alpha, bp, bm);
      a[H + n] = t;
      regsum += fabsf(t);
    }
  }
  __syncthreads();
}

template <int NB>
__device__ void inv_level_wmma(float* a, int H, const float* bt0, const float* bt1, int tid) {
  const int mask = H - 1;
  const int wave = tid >> 5;
  const int lane = tid & 31;
  const int lo = lane & 15;
  const int hi = lane >> 4;
  const int kloc = hi ? 2 : 0;

  v2f Bsc[6], Bws[6];  // bt0: scaling (even_rest), bt1: ws (odd_rest)
#pragma unroll
  for (int s = 0; s < 6; s++) {
    int r = 4 * s + kloc;
    Bsc[s] = v2f{bt0[r * 16 + lo], bt0[(r + 1) * 16 + lo]};
    Bws[s] = v2f{bt1[r * 16 + lo], bt1[(r + 1) * 16 + lo]};
  }

  v8f accEr[NB], accOr[NB];
#pragma unroll
  for (int b = 0; b < NB; b++) {
    const int n0 = (wave + 32 * b) * 256;
    const int base = n0 + 16 * lo;  // correlation window (n+k)
    v8f aco = {0.f, 0.f, 0.f, 0.f, 0.f, 0.f, 0.f, 0.f};
    v8f ace = {0.f, 0.f, 0.f, 0.f, 0.f, 0.f, 0.f, 0.f};
#pragma unroll
    for (int s = 0; s < 6; s++) {
      int m0 = 4 * s + kloc;
      int i0 = (base + m0) & mask;
      int i1 = (base + m0 + 1) & mask;
      v2f Aap = v2f{a[i0],     a[i1]};      // approx window
      v2f Adt = v2f{a[H + i0], a[H + i1]};  // details window
      aco = __builtin_amdgcn_wmma_f32_16x16x4_f32(false, Aap, false, Bsc[s],
                                                  (short)0, aco, false, false);
      ace = __builtin_amdgcn_wmma_f32_16x16x4_f32(false, Adt, false, Bws[s],
                                                  (short)0, ace, false, false);
    }
    v8f dv, av;
#pragma unroll
    for (int v = 0; v < 8; v++) {
      int n = n0 + 16 * (v + 8 * hi) + lo;
      dv[v] = a[H + n];
      av[v] = a[n];
    }
    accEr[b] = dv + aco;  // even_rest = details + corr(approx, scaling)
    accOr[b] = av + ace;  // odd_rest  = approx  + corr(details, ws)
  }
  __syncthreads();
#pragma unroll
  for (int b = 0; b < NB; b++) {
    const int n0 = (wave + 32 * b) * 256;
#pragma unroll
    for (int v = 0; v < 8; v++) {
      int n = n0 + 16 * (v + 8 * hi) + lo;
      a[2 * n]     = accEr[b][v];
      a[2 * n + 1] = accOr[b][v];
    }
  }
  __syncthreads();
}

#endif  // HAVE_WMMA_F32

__global__ __launch_bounds__(TPB)
void wavelet_row_kernel(const float* __restrict__ x,
                        const float* __restrict__ sc,
                        const float* __restrict__ sr,
                        const float* __restrict__ alpha_p,
                        const float* __restrict__ bp_p,
                        const float* __restrict__ bm_p,
                        float* __restrict__ out,
                        float* __restrict__ partial) {
  __shared__ float buf[ROW];     // 256 KB: whole row resident (CDNA5 320KB LDS/WGP)
  __shared__ float red[TPB];
  __shared__ float bt0[24 * 16]; // Toeplitz band tables for WMMA levels
  __shared__ float bt1[24 * 16];

  const int row = blockIdx.x;
  const int tid = threadIdx.x;
  const float alpha = *alpha_p, bp = *bp_p, bm = *bm_p;

  // ---- load row into LDS (async global->LDS on gfx1250, ASYNCcnt path) ----
  {
#if HAVE_ASYNC_LDS
    char* gsrc = (char*)(x + (size_t)row * ROW);
    char* lbase = (char*)buf;
#pragma unroll
    for (int c = 0; c < (ROW / 4) / TPB; c++) {
      int idx = c * TPB + tid;  // 16-byte chunks
      __builtin_amdgcn_global_load_async_to_lds_b128(
          (gv4i_ptr)(gsrc + (size_t)idx * 16),
          (lv4i_ptr)(lbase + (size_t)idx * 16),
          0, 0);
    }
#if __has_builtin(__builtin_amdgcn_s_wait_asynccnt)
    __builtin_amdgcn_s_wait_asynccnt(0);
#else
    asm volatile("s_wait_asynccnt 0" ::: "memory");
#endif
#else
    const float4* x4 = (const float4*)(x + (size_t)row * ROW);
    float4* b4 = (float4*)buf;
#pragma unroll
    for (int c = 0; c < (ROW / 4) / TPB; c++) {
      int idx = c * TPB + tid;
      b4[idx] = x4[idx];
    }
#endif
  }
  __syncthreads();

  float regsum = 0.0f;

#if HAVE_WMMA_F32
  // Toeplitz tables:
  //  fwd: bt0[m][j] = wav[7-(m-j)]     for 0<=m-j<8 else 0  (circconv, n-k)
  //       bt1[m][j] = scaling[7-(m-j)]
  //  inv: bt0[m][j] = scaling[m-j]     for 0<=m-j<8 else 0  (correlation, n+k)
  //       bt1[m][j] = ws[m-j]
  auto fill_fwd_tabs = [&](int lvl) {
    if (tid < 24 * 16) {
      int m = tid >> 4, j = tid & 15, d = m - j;
      float v0 = 0.f, v1 = 0.f;
      if (d >= 0 && d < 8) {
        float srd = sr[lvl * 8 + d];               // wav[7-d] = +/- s_rec[d]
        v0 = ((7 - d) & 1) ? -srd : srd;
        v1 = sc[lvl * 8 + 7 - d];                  // scaling[7-d]
      }
      bt0[tid] = v0;
      bt1[tid] = v1;
    }
  };
  auto fill_inv_tabs = [&](int lvl) {
    if (tid < 24 * 16) {
      int m = tid >> 4, j = tid & 15, d = m - j;
      float v0 = 0.f, v1 = 0.f;
      if (d >= 0 && d < 8) {
        v0 = sc[lvl * 8 + d];                      // scaling[d]
        float g = sr[lvl * 8 + 7 - d];             // ws[d] = (d odd ? + : -) s_rec[7-d]
        v1 = (d & 1) ? g : -g;
      }
      bt0[tid] = v0;
      bt1[tid] = v1;
    }
  };
#endif

  // ---- forward lifting: level l works on a[0 .. 2H_l), H_l = 32768 >> l ----
#define FWD(LVL, CPAR)                                                   \
  { float sf_[8], wf_[8];                                                \
    ld_fwd_filters(sc, sr, LVL, sf_, wf_);                               \
    fwd_level<CPAR>(buf, (ROW / 2) >> (LVL), wf_, sf_, alpha, bp, bm,    \
                    regsum, tid); }
#if HAVE_WMMA_F32
  fill_fwd_tabs(0); __syncthreads();
  fwd_level_wmma<4>(buf, 32768, bt0, bt1, alpha, bp, bm, regsum, tid);
  fill_fwd_tabs(1); __syncthreads();
  fwd_level_wmma<2>(buf, 16384, bt0, bt1, alpha, bp, bm, regsum, tid);
  fill_fwd_tabs(2); __syncthreads();
  fwd_level_wmma<1>(buf, 8192,  bt0, bt1, alpha, bp, bm, regsum, tid);
#else
  FWD(0, 32) FWD(1, 16) FWD(2, 8)
#endif
  FWD(3, 4) FWD(4, 2) FWD(5, 1) FWD(6, 1) FWD(7, 1) FWD(8, 1) FWD(9, 1)
#undef FWD

  // ---- threshold the final 64-element approx ----
  if (tid < 64) {
    float t = thresh(buf[tid], alpha, bp, bm);
    buf[tid] = t;
    regsum += fabsf(t);
  }
  __syncthreads();

  // ---- inverse lifting: level l rebuilds a[0 .. 2H_l) from halves ----
#define INV(LVL, CPAR)                                                   \
  { float sf_[8], ws_[8];                                                \
    ld_inv_filters(sc, sr, LVL, sf_, ws_);                               \
    inv_level<CPAR>(buf, (ROW / 2) >> (LVL), sf_, ws_, tid); }
  INV(9, 1) INV(8, 1) INV(7, 1) INV(6, 1) INV(5, 1)
  INV(4, 2) INV(3, 4)
#if HAVE_WMMA_F32
  fill_inv_tabs(2); __syncthreads();
  inv_level_wmma<1>(buf, 8192,  bt0, bt1, tid);
  fill_inv_tabs(1); __syncthreads();
  inv_level_wmma<2>(buf, 16384, bt0, bt1, tid);
  fill_inv_tabs(0); __syncthreads();
  inv_level_wmma<4>(buf, 32768, bt0, bt1, tid);
#else
  INV(2, 8) INV(1, 16) INV(0, 32)
#endif
#undef INV

  // ---- store reconstructed row ----
  {
    float4* o4 = (float4*)(out + (size_t)row * ROW);
    const float4* b4 = (const float4*)buf;
#pragma unroll
    for (int c = 0; c < (ROW / 4) / TPB; c++) {
      int idx = c * TPB + tid;
      o4[idx] = b4[idx];
    }
  }

  // ---- per-row reduction of reg-loss partial (deterministic tree) ----
  red[tid] = regsum;
  __syncthreads();
  for (int s = TPB / 2; s > 0; s >>= 1) {
    if (tid < s) red[tid] += red[tid + s];
    __syncthreads();
  }
  if (tid == 0) partial[row] = red[0];
}

__global__ void reduce_loss_kernel(const float* __restrict__ partial, int nrows,
                                   float* __restrict__ out_loss) {
  __shared__ float red[256];
  int t = threadIdx.x;
  red[t] = (t < nrows) ? partial[t] : 0.0f;
  __syncthreads();
  for (int s = 128; s > 0; s >>= 1) {
    if (t < s) red[t] += red[t + s];
    __syncthreads();
  }
  if (t == 0) out_loss[0] = red[0];
}

extern "C" void kernel_launch(void* const* d_in, const int* in_sizes, int n_in,
                              void* d_out, int out_size, void* d_ws, size_t ws_size,
                              hipStream_t stream) {
  const float* x  = (const float*)d_in[0];
  const float* sc = (const float*)d_in[1];  // scaling      (LEVELS, 8)
  const float* sr = (const float*)d_in[2];  // scaling_rec  (LEVELS, 8)
  const float* al = (const float*)d_in[3];
  const float* bp = (const float*)d_in[4];
  const float* bm = (const float*)d_in[5];
  float* out = (float*)d_out;
  float* partial = (float*)d_ws;

  const int nrows = in_sizes[0] / ROW;  // 256

  wavelet_row_kernel<<<nrows, TPB, 0, stream>>>(x, sc, sr, al, bp, bm, out, partial);
  reduce_loss_kernel<<<1, 256, 0, stream>>>(partial, nrows,
                                            out + (size_t)nrows * ROW);
}